// TNModel_43001212568020
// MI455X (gfx1250) — compile-verified
//
#include <hip/hip_runtime.h>

typedef __attribute__((ext_vector_type(2))) float v2f;
typedef __attribute__((ext_vector_type(4))) float v4f;
typedef __attribute__((ext_vector_type(8))) float v8f;

// D = A(16x4 f32) * B(4x16 f32) + C(16x16 f32), wave32 CDNA5 WMMA
#define WMMA4(a, b, c) \
  __builtin_amdgcn_wmma_f32_16x16x4_f32(false, (a), false, (b), (short)0, (c), false, false)

// Load 8 consecutive floats (two b128) into a register array.
__device__ __forceinline__ void ld8(const float* p, float* d) {
  v4f a = *(const v4f*)(p);
  v4f b = *(const v4f*)(p + 4);
  d[0] = a.x; d[1] = a.y; d[2] = a.z; d[3] = a.w;
  d[4] = b.x; d[5] = b.y; d[6] = b.z; d[7] = b.w;
}

// Batch-load all B fragments of one node into registers (one clause + 1 wait).
// cc is the N column CLAMPED to DO-1: loads are unconditional and in-bounds;
// lanes col >= DO get duplicate finite values that only reach D columns which
// are never stored or consumed.
template <int DI, int DO>
__device__ __forceinline__ void load_b(const float* __restrict__ Wn,
                                       int hi, int cc, v2f* bb) {
#pragma unroll
  for (int c = 0; c < (DI * DI) / 4; ++c) {
    const float* wr = Wn + (4 * c + 2 * hi) * DO + cc;
    bb[c].x = wr[0]; bb[c].y = wr[DO];
  }
}

// Two interleaved accumulator chains (two 16-sample tiles) sharing B:
//   D[m,k] = sum_{ij} (L[m,i]*R[m,j]) * W[n,i,j,k]
// A-chunk c covers ij = 4c..4c+3; per-lane (m = lane%16, hi = lane/16)
// A vgpr p holds O[m, 4c + 2*hi + p]: i compile-time, j a 2-way select.
template <int DI, int DO>
__device__ __forceinline__ void tn_mma2(const float* L0, const float* R0,
                                        const float* L1, const float* R1,
                                        const v2f* bb, int hi,
                                        v8f& acc0, v8f& acc1) {
#pragma unroll
  for (int c = 0; c < (DI * DI) / 4; ++c) {
    const int i  = (4 * c) / DI;   // constant within a chunk (DI = 4 or 8)
    const int j0 = (4 * c) % DI;   // 0 or 4
    v2f a0; a0.x = L0[i] * (hi ? R0[j0 + 2] : R0[j0 + 0]);
            a0.y = L0[i] * (hi ? R0[j0 + 3] : R0[j0 + 1]);
    acc0 = WMMA4(a0, bb[c], acc0);
    v2f a1; a1.x = L1[i] * (hi ? R1[j0 + 2] : R1[j0 + 0]);
            a1.y = L1[i] * (hi ? R1[j0 + 3] : R1[j0 + 1]);
    acc1 = WMMA4(a1, bb[c], acc1);
  }
}

// One tree level for both tiles: NOUT parent nodes; children read from LDS
// (8-float slots, padded per-sample stride, 32 sample rows), D written back
// in tile layout.  B fragments loaded once, shared by both tiles.
template <int DI, int NOUT, int DO>
__device__ __forceinline__ void tn_level(const float* __restrict__ W, int wbase,
                                         const float* sin, int istr,
                                         float* sout, int ostr,
                                         int m, int hi, int col) {
  const bool cv = col < DO;
  const int  cc = cv ? col : (DO - 1);
  for (int p = 0; p < NOUT; ++p) {
    v2f bb[(DI * DI) / 4];
    load_b<DI, DO>(W + (size_t)(wbase + p) * (DI * DI * DO), hi, cc, bb);
    float L0[8], R0[8], L1[8], R1[8];
    ld8(sin + m * istr + (2 * p) * 8, L0);
    ld8(sin + m * istr + (2 * p + 1) * 8, R0);
    ld8(sin + (16 + m) * istr + (2 * p) * 8, L1);
    ld8(sin + (16 + m) * istr + (2 * p + 1) * 8, R1);
    v8f acc0 = {}, acc1 = {};
    tn_mma2<DI, DO>(L0, R0, L1, R1, bb, hi, acc0, acc1);
#pragma unroll
    for (int r = 0; r < 8; ++r) {
      if (cv) {
        sout[(r + 8 * hi) * ostr + p * 8 + col]        = acc0[r];
        sout[(16 + r + 8 * hi) * ostr + p * 8 + col]   = acc1[r];
      }
    }
  }
}

__global__ __launch_bounds__(32) void tn_tree_kernel(
    const float* __restrict__ x,
    const float* __restrict__ w7, const float* __restrict__ w6,
    const float* __restrict__ w5, const float* __restrict__ w4,
    const float* __restrict__ w3, const float* __restrict__ w2,
    const float* __restrict__ w1, const float* __restrict__ w0,
    float* __restrict__ out) {
  // Per-sample strides padded by 4 floats -> conflict-free b128 LDS loads.
  constexpr int SA = 16 * 8 + 4;  // leaf output, 16 nodes/subtree
  constexpr int SB = 8 * 8 + 4;   // w6 output, 8 nodes
  constexpr int SC = 4 * 8 + 4;   // w5 / w2 output, 4 nodes
  constexpr int SD = 2 * 8 + 4;   // w4 / w1 output, 2 nodes
  constexpr int SK = 8 * 8 + 4;   // w3 collect across 8 subtrees

  __shared__ float sA[32 * SA];   // 32 sample rows = two 16-sample tiles
  __shared__ float sB[32 * SB];
  __shared__ float sC[32 * SC];
  __shared__ float sD[32 * SD];
  __shared__ float sK[32 * SK];
  __shared__ float sE[32 * SC];
  __shared__ float sF[32 * SD];

  const int l   = threadIdx.x;
  const int m   = l & 15;   // sample row within a 16-sample tile
  const int hi  = l >> 4;   // half-wave: K sub-tile (A/B) / M-row half (D) select
  const int col = m;        // N index = lane % 16
  const int b0  = blockIdx.x * 32;

#pragma unroll 1
  for (int st = 0; st < 8; ++st) {  // 8 subtrees of 32 features each
    {  // leaf layer: d_in=2, d_out=4, 16 nodes per subtree; x from global
      const bool cv = col < 4;
      const int  cc = cv ? col : 3;  // clamp -> unconditional weight loads
      const float* xs0 = x + ((size_t)(b0 + m) * 256 + st * 32) * 2;
      const float* xs1 = xs0 + 16 * 512;  // tile 1: samples b0+16..b0+31
      for (int nl = 0; nl < 16; ++nl) {
        const float* wn = w7 + (size_t)(st * 16 + nl) * 16 + (2 * hi) * 4 + cc;
        v2f b; b.x = wn[0]; b.y = wn[4];
        v4f xv0 = *(const v4f*)(xs0 + nl * 4);  // [L0 L1 R0 R1], 16B aligned
        v4f xv1 = *(const v4f*)(xs1 + nl * 4);
        v2f a0; { float Ls = hi ? xv0.y : xv0.x; a0.x = Ls * xv0.z; a0.y = Ls * xv0.w; }
        v2f a1; { float Ls = hi ? xv1.y : xv1.x; a1.x = Ls * xv1.z; a1.y = Ls * xv1.w; }
        v8f acc0 = {}, acc1 = {};
        acc0 = WMMA4(a0, b, acc0);
        acc1 = WMMA4(a1, b, acc1);
#pragma unroll
        for (int r = 0; r < 8; ++r) {
          if (cv) {
            sA[(r + 8 * hi) * SA + nl * 8 + col]      = acc0[r];
            sA[(16 + r + 8 * hi) * SA + nl * 8 + col] = acc1[r];
          }
        }
      }
    }
    __syncthreads();
    tn_level<4, 8, 8>(w6, st * 8, sA, SA, sB, SB, m, hi, col);
    __syncthreads();
    tn_level<8, 4, 8>(w5, st * 4, sB, SB, sC, SC, m, hi, col);
    __syncthreads();
    tn_level<8, 2, 8>(w4, st * 2, sC, SC, sD, SD, m, hi, col);
    __syncthreads();
    tn_level<8, 1, 8>(w3, st, sD, SD, sK + st * 8, SK, m, hi, col);
    __syncthreads();
  }

  tn_level<8, 4, 8>(w2, 0, sK, SK, sE, SC, m, hi, col);
  __syncthreads();
  tn_level<8, 2, 8>(w1, 0, sE, SC, sF, SD, m, hi, col);
  __syncthreads();

  {  // root: d_in=8, d_out=2 -> global output [B, 2], both tiles
    const bool cv = col < 2;
    const int  cc = cv ? col : 1;
    v2f bb[16];
    load_b<8, 2>(w0, hi, cc, bb);
    float L0[8], R0[8], L1[8], R1[8];
    ld8(sF + m * SD + 0, L0);
    ld8(sF + m * SD + 8, R0);
    ld8(sF + (16 + m) * SD + 0, L1);
    ld8(sF + (16 + m) * SD + 8, R1);
    v8f acc0 = {}, acc1 = {};
    tn_mma2<8, 2>(L0, R0, L1, R1, bb, hi, acc0, acc1);
#pragma unroll
    for (int r = 0; r < 8; ++r) {
      if (cv) {
        out[(size_t)(b0 + r + 8 * hi) * 2 + col]      = acc0[r];
        out[(size_t)(b0 + 16 + r + 8 * hi) * 2 + col] = acc1[r];
      }
    }
  }
}

extern "C" void kernel_launch(void* const* d_in, const int* in_sizes, int n_in,
                              void* d_out, int out_size, void* d_ws, size_t ws_size,
                              hipStream_t stream) {
  const float* x  = (const float*)d_in[0];
  const float* w7 = (const float*)d_in[1];
  const float* w6 = (const float*)d_in[2];
  const float* w5 = (const float*)d_in[3];
  const float* w4 = (const float*)d_in[4];
  const float* w3 = (const float*)d_in[5];
  const float* w2 = (const float*)d_in[6];
  const float* w1 = (const float*)d_in[7];
  const float* w0 = (const float*)d_in[8];
  float* out = (float*)d_out;
  (void)in_sizes; (void)n_in; (void)out_size; (void)d_ws; (void)ws_size;

  dim3 grid(65536 / 32);  // one wave32 per two 16-sample tiles
  dim3 block(32);
  tn_tree_kernel<<<grid, block, 0, stream>>>(x, w7, w6, w5, w4, w3, w2, w1, w0, out);
}